// BinaryTreeLSTM_90632399880487
// MI455X (gfx1250) — compile-verified
//
#include <hip/hip_runtime.h>

typedef __attribute__((ext_vector_type(2))) float v2f;
typedef __attribute__((ext_vector_type(8))) float v8f;

// D = A(16x4 f32) x B(4x16 f32) + C(16x16 f32), full-precision CDNA5 matrix op.
#define WMMA_F32(a, b, c) \
    __builtin_amdgcn_wmma_f32_16x16x4_f32(false, (a), false, (b), (short)0, (c), false, false)

__device__ __forceinline__ float sigmoid_f(float x) {
    return 1.0f / (1.0f + __expf(-x));
}

// ---------------------------------------------------------------------------
// Leaf phase: c = in @ Wcx^T + bcx ; h = sigmoid(in @ Wox^T + box) * tanh(c)
// One wave = 16 rows x 128 cols. A panel (16x128) held in 64 VGPRs.
// ---------------------------------------------------------------------------
__global__ void __launch_bounds__(256) leaf_kernel(
    const float* __restrict__ in,
    const float* __restrict__ Wcx, const float* __restrict__ bcx,
    const float* __restrict__ Wox, const float* __restrict__ box,
    float* __restrict__ c_out, float* __restrict__ h_out, int nrows)
{
    const int lane  = threadIdx.x & 31;
    const int wave  = threadIdx.x >> 5;
    const int panel = blockIdx.x * 8 + wave;
    const int row0  = panel * 16;
    if (row0 >= nrows) return;           // wave-uniform
    const int half = lane >> 4;          // 0: lanes 0-15, 1: lanes 16-31
    const int lm   = lane & 15;

    // A-matrix layout (ISA 7.12.2, 32-bit A 16x4): lane m holds row M=m;
    // VGPR0/1 = K = k0+2*half, k0+1+2*half.
    v2f A[32];
    {
        const float* ap = in + (size_t)(row0 + lm) * 128 + 2 * half;
        #pragma unroll
        for (int k = 0; k < 32; ++k) A[k] = *(const v2f*)(ap + 4 * k);
    }

    for (int t = 0; t < 8; ++t) {
        const int col0 = t * 16;
        const float bc = bcx[col0 + lm];
        const float bo = box[col0 + lm];
        v8f accc = {bc, bc, bc, bc, bc, bc, bc, bc};
        v8f acco = {bo, bo, bo, bo, bo, bo, bo, bo};
        // B[k][n] = W[n][k]; lane n holds col N=n, VGPR0/1 = K rows (same split as A).
        const float* wc = Wcx + (size_t)(col0 + lm) * 128 + 2 * half;
        const float* wo = Wox + (size_t)(col0 + lm) * 128 + 2 * half;
        #pragma unroll
        for (int k = 0; k < 32; ++k) {
            v2f bC = *(const v2f*)(wc + 4 * k);
            v2f bO = *(const v2f*)(wo + 4 * k);
            accc = WMMA_F32(A[k], bC, accc);
            acco = WMMA_F32(A[k], bO, acco);
        }
        // C/D layout: VGPR j -> M = j + 8*half, N = lane%16
        #pragma unroll
        for (int j = 0; j < 8; ++j) {
            const int m = j + 8 * half;
            const float cv = accc[j];
            const float hv = sigmoid_f(acco[j]) * tanhf(cv);
            const size_t idx = (size_t)(row0 + m) * 128 + col0 + lm;
            c_out[idx] = cv;
            h_out[idx] = hv;
        }
    }
}

// ---------------------------------------------------------------------------
// Tree level: X pairs. g[gi] = lh@Wl[gi]^T + rh@Wr[gi]^T + (bl+br)[gi]
// c = sig(g0)*tanh(g4) + sig(g1)*lc + sig(g2)*rc ; h = sig(g3)*tanh(c)
// One wave = 16 pairs x 128 mem dims; lh/rh panels in 128 VGPRs; weights
// streamed from L2 (640 KB total, fully L2-resident).
// ---------------------------------------------------------------------------
__global__ void __launch_bounds__(256) tree_kernel(
    const float* __restrict__ c_in, const float* __restrict__ h_in,
    const float* __restrict__ Wl, const float* __restrict__ bl,
    const float* __restrict__ Wr, const float* __restrict__ br,
    float* __restrict__ c_out, float* __restrict__ h_out, int X)
{
    const int lane  = threadIdx.x & 31;
    const int wave  = threadIdx.x >> 5;
    const int panel = blockIdx.x * 8 + wave;
    const int p0    = panel * 16;
    if (p0 >= X) return;                 // wave-uniform
    const int half = lane >> 4;
    const int lm   = lane & 15;

    int pr = p0 + lm;
    pr = (pr < X) ? pr : (X - 1);        // per-lane clamp (select, no EXEC change)

    v2f AL[32], AR[32];
    {
        const float* lp = h_in + (size_t)(2 * pr) * 128 + 2 * half;
        const float* rp = h_in + (size_t)(2 * pr + 1) * 128 + 2 * half;
        #pragma unroll
        for (int k = 0; k < 32; ++k) {
            AL[k] = *(const v2f*)(lp + 4 * k);
            AR[k] = *(const v2f*)(rp + 4 * k);
        }
    }

    for (int t = 0; t < 8; ++t) {
        const int col0 = t * 16;
        v8f g[5];
        #pragma unroll
        for (int gi = 0; gi < 5; ++gi) {
            const float b = bl[gi * 128 + col0 + lm] + br[gi * 128 + col0 + lm];
            v8f tmp = {b, b, b, b, b, b, b, b};
            g[gi] = tmp;
        }
        #pragma unroll
        for (int gi = 0; gi < 5; ++gi) {
            const float* wl = Wl + (size_t)gi * 16384 + (size_t)(col0 + lm) * 128 + 2 * half;
            const float* wr = Wr + (size_t)gi * 16384 + (size_t)(col0 + lm) * 128 + 2 * half;
            __builtin_prefetch(wl, 0, 1);    // global_prefetch_b8 into L2
            __builtin_prefetch(wr, 0, 1);
            #pragma unroll
            for (int k = 0; k < 32; ++k) {
                v2f bL = *(const v2f*)(wl + 4 * k);
                v2f bR = *(const v2f*)(wr + 4 * k);
                v8f a = g[gi];
                a = WMMA_F32(AL[k], bL, a);
                a = WMMA_F32(AR[k], bR, a);
                g[gi] = a;
            }
        }
        #pragma unroll
        for (int j = 0; j < 8; ++j) {
            const int p = p0 + j + 8 * half;
            if (p < X) {
                const size_t base = (size_t)(2 * p) * 128 + col0 + lm;
                const float lc = c_in[base];
                const float rc = c_in[base + 128];
                const float iv = sigmoid_f(g[0][j]);
                const float lf = sigmoid_f(g[1][j]);
                const float rf = sigmoid_f(g[2][j]);
                const float ov = sigmoid_f(g[3][j]);
                const float uv = tanhf(g[4][j]);
                const float cv = iv * uv + lf * lc + rf * rc;
                const size_t oi = (size_t)p * 128 + col0 + lm;
                c_out[oi] = cv;
                h_out[oi] = ov * tanhf(cv);
            }
        }
    }
}

// ---------------------------------------------------------------------------
extern "C" void kernel_launch(void* const* d_in, const int* in_sizes, int n_in,
                              void* d_out, int out_size, void* d_ws, size_t ws_size,
                              hipStream_t stream) {
    const float* inputs = (const float*)d_in[0];
    const float* Wcx    = (const float*)d_in[1];
    const float* bcx    = (const float*)d_in[2];
    const float* Wox    = (const float*)d_in[3];
    const float* box    = (const float*)d_in[4];
    const float* Wl     = (const float*)d_in[5];
    const float* bl     = (const float*)d_in[6];
    const float* Wr     = (const float*)d_in[7];
    const float* br     = (const float*)d_in[8];
    float* out = (float*)d_out;

    const int N = in_sizes[0] / 128;           // 262144 leaves
    float* ws = (float*)d_ws;
    const size_t cap = (size_t)N * 128;
    // Ping-pong activation buffers: A holds up to N rows, B up to N/2 rows.
    float* Ac = ws;
    float* Ah = ws + cap;
    float* Bc = ws + 2 * cap;
    float* Bh = ws + 2 * cap + cap / 2;

    {
        const int panels = (N + 15) / 16;
        const int blocks = (panels + 7) / 8;
        leaf_kernel<<<blocks, 256, 0, stream>>>(inputs, Wcx, bcx, Wox, box, Ac, Ah, N);
    }

    const float* cc  = Ac;
    const float* chh = Ah;
    bool curA = true;
    int rows = N;
    while (rows > 1) {
        const int X = rows >> 1;
        float *oc, *oh;
        if (X == 1)      { oc = out; oh = out + 128; }   // final (c, h) -> d_out
        else if (curA)   { oc = Bc;  oh = Bh; }
        else             { oc = Ac;  oh = Ah; }
        const int panels = (X + 15) / 16;
        const int blocks = (panels + 7) / 8;
        tree_kernel<<<blocks, 256, 0, stream>>>(cc, chh, Wl, bl, Wr, br, oc, oh, X);
        cc = oc; chh = oh;
        curA = !curA;
        rows = X;
    }
}